// SAGE_7026566496443
// MI455X (gfx1250) — compile-verified
//
#include <hip/hip_runtime.h>
#include <hip/hip_bf16.h>

#define DFEAT 64

typedef __attribute__((ext_vector_type(2))) float v2f;
typedef __attribute__((ext_vector_type(8))) float v8f;

// ---------------------------------------------------------------- utilities
__global__ void zero_f32_kernel(float* __restrict__ p, size_t n) {
    size_t i = (size_t)blockIdx.x * blockDim.x + threadIdx.x;
    size_t stride = (size_t)gridDim.x * blockDim.x;
    for (; i < n; i += stride) p[i] = 0.0f;
}

__global__ void deg_count_kernel(const int* __restrict__ dst, float* __restrict__ deg, int E) {
    int e = blockIdx.x * blockDim.x + threadIdx.x;
    if (e < E) atomicAdd(&deg[dst[e]], 1.0f);
}

__global__ void inv_deg_kernel(float* __restrict__ deg, int n) {
    int i = blockIdx.x * blockDim.x + threadIdx.x;
    if (i < n) deg[i] = 1.0f / fmaxf(deg[i], 1.0f);
}

// ------------------------------------------------------- edge scatter (mean num.)
// One thread per (edge, 4-float chunk): float4 gather of source row,
// 4x global_atomic_add_f32 into agg[dst]. idx space = E*16 (25.6M, fits int).
__global__ void scatter_add_kernel(const float* __restrict__ hin,
                                   const int* __restrict__ src,
                                   const int* __restrict__ dst,
                                   float* __restrict__ agg, int E) {
    int idx = blockIdx.x * blockDim.x + threadIdx.x;
    int total = E * 16;
    if (idx >= total) return;
    int e = idx >> 4;
    int c = (idx & 15) << 2;
    int s = src[e];
    int d = dst[e];
    const float4 v = *(const float4*)(hin + (size_t)s * DFEAT + c);
    float* p = agg + (size_t)d * DFEAT + c;
    atomicAdd(p + 0, v.x);
    atomicAdd(p + 1, v.y);
    atomicAdd(p + 2, v.z);
    atomicAdd(p + 3, v.w);
}

// ------------------------------------------------------- fused SAGE GEMM (WMMA f32)
// out[16t..16t+15, :] = relu?( (agg*invdeg) @ Wl + hin @ Wr + b )
// One wave32 per 16-row node tile; V_WMMA_F32_16X16X4_F32, K stepped by 4.
// A frag (16x4 f32): lane m = lane&15, koff = (lane>>4)*2; v0=A[m][k+koff], v1=A[m][k+koff+1]
// B frag (4x16 f32): lane n = lane&15 (col), same koff K split.
// C/D (16x16 f32, 8 VGPRs): vgpr r holds row r (lanes 0-15) / r+8 (lanes 16-31).
//
// Weights are staged in LDS packed as [K/2][64][2] so each lane's B-fragment
// pair (W[kb][n], W[kb+1][n]) is one 8-byte ds_load_b64 into an even VGPR pair
// (no register shuffles; K-step addresses fold into DS immediate offsets).
__launch_bounds__(256)
__global__ void sage_gemm_wmma_kernel(const float* __restrict__ agg,
                                      const float* __restrict__ hin,
                                      const float* __restrict__ Wl,
                                      const float* __restrict__ Wr,
                                      const float* __restrict__ bias,
                                      const float* __restrict__ invdeg,
                                      float* __restrict__ out,
                                      int n_tiles, int do_relu) {
    __shared__ __align__(16) float sWl[DFEAT * DFEAT];   // packed [32][64][2]
    __shared__ __align__(16) float sWr[DFEAT * DFEAT];   // packed [32][64][2]
    __shared__ float sB[DFEAT];

    for (int i = threadIdx.x; i < DFEAT * DFEAT; i += blockDim.x) {
        const int k = i >> 6;
        const int n = i & 63;
        const int p = ((k >> 1) << 7) + (n << 1) + (k & 1);
        sWl[p] = Wl[i];
        sWr[p] = Wr[i];
    }
    if (threadIdx.x < DFEAT) sB[threadIdx.x] = bias[threadIdx.x];
    __syncthreads();

    const int wave = threadIdx.x >> 5;
    const int lane = threadIdx.x & 31;
    const int tile = blockIdx.x * 8 + wave;
    if (tile >= n_tiles) return;   // wave-uniform: EXEC stays all-1s for WMMA

    const int m    = lane & 15;
    const int koff = (lane >> 4) << 1;          // 0 or 2
    const int row  = tile * 16 + m;
    const float inv = invdeg[row];

    // Load all A fragments once (16 K-steps x 2 matrices)
    v2f aA[16], aX[16];
    const float* pa = agg + (size_t)row * DFEAT + koff;
    const float* px = hin + (size_t)row * DFEAT + koff;
#pragma unroll
    for (int k = 0; k < 16; ++k) {
        v2f t = *(const v2f*)(pa + 4 * k);
        aA[k].x = t.x * inv;
        aA[k].y = t.y * inv;
        aX[k] = *(const v2f*)(px + 4 * k);
    }

#pragma unroll
    for (int j = 0; j < 4; ++j) {               // 4 tiles of 16 output columns
        v8f c = {};
        const int n = (j << 4) + m;
#pragma unroll
        for (int k = 0; k < 16; ++k) {
            const int kb = (k << 2) + koff;                   // even
            const int pbase = ((kb >> 1) << 7) + (n << 1);    // packed index
            v2f bl = *(const v2f*)(sWl + pbase);
            v2f br = *(const v2f*)(sWr + pbase);
            c = __builtin_amdgcn_wmma_f32_16x16x4_f32(false, aA[k], false, bl,
                                                      (short)0, c, false, false);
            c = __builtin_amdgcn_wmma_f32_16x16x4_f32(false, aX[k], false, br,
                                                      (short)0, c, false, false);
        }
        const float bv = sB[n];
        const int rbase = tile * 16 + ((lane >> 4) << 3);
#pragma unroll
        for (int r = 0; r < 8; ++r) {
            float v = c[r] + bv;
            if (do_relu) v = fmaxf(v, 0.0f);
            out[(size_t)(rbase + r) * DFEAT + n] = v;
        }
    }
}

// ---------------------------------------------------------------- launch
extern "C" void kernel_launch(void* const* d_in, const int* in_sizes, int n_in,
                              void* d_out, int out_size, void* d_ws, size_t ws_size,
                              hipStream_t stream) {
    const float* x   = (const float*)d_in[0];
    const float* Wl1 = (const float*)d_in[1];
    const float* Wr1 = (const float*)d_in[2];
    const float* b1  = (const float*)d_in[3];
    const float* Wl2 = (const float*)d_in[4];
    const float* Wr2 = (const float*)d_in[5];
    const float* b2  = (const float*)d_in[6];
    const float* Wl3 = (const float*)d_in[7];
    const float* Wr3 = (const float*)d_in[8];
    const float* b3  = (const float*)d_in[9];
    const int* esrc  = (const int*)d_in[10];
    const int* edst  = (const int*)d_in[11];

    const int E = in_sizes[10];
    const int N = in_sizes[0] / DFEAT;          // 100000, divisible by 16

    float* deg = (float*)d_ws;                  // N
    float* agg = deg + N;                       // N*64
    float* h1  = agg + (size_t)N * DFEAT;       // N*64
    float* h2  = h1 + (size_t)N * DFEAT;        // N*64
    float* outp = (float*)d_out;

    const int n_tiles = (N + 15) / 16;
    dim3 gemm_grid((n_tiles + 7) / 8);
    const int scat_blocks = (E * 16 + 255) / 256;

    // degree (computed fresh every call: deterministic, capture-safe)
    zero_f32_kernel<<<1024, 256, 0, stream>>>(deg, (size_t)N);
    deg_count_kernel<<<(E + 255) / 256, 256, 0, stream>>>(edst, deg, E);
    inv_deg_kernel<<<(N + 255) / 256, 256, 0, stream>>>(deg, N);

    // layer 1: x -> h1 (relu)
    zero_f32_kernel<<<4096, 256, 0, stream>>>(agg, (size_t)N * DFEAT);
    scatter_add_kernel<<<scat_blocks, 256, 0, stream>>>(x, esrc, edst, agg, E);
    sage_gemm_wmma_kernel<<<gemm_grid, 256, 0, stream>>>(agg, x, Wl1, Wr1, b1, deg, h1,
                                                         n_tiles, 1);

    // layer 2: h1 -> h2 (relu)
    zero_f32_kernel<<<4096, 256, 0, stream>>>(agg, (size_t)N * DFEAT);
    scatter_add_kernel<<<scat_blocks, 256, 0, stream>>>(h1, esrc, edst, agg, E);
    sage_gemm_wmma_kernel<<<gemm_grid, 256, 0, stream>>>(agg, h1, Wl2, Wr2, b2, deg, h2,
                                                         n_tiles, 1);

    // layer 3: h2 -> out (no relu)
    zero_f32_kernel<<<4096, 256, 0, stream>>>(agg, (size_t)N * DFEAT);
    scatter_add_kernel<<<scat_blocks, 256, 0, stream>>>(h2, esrc, edst, agg, E);
    sage_gemm_wmma_kernel<<<gemm_grid, 256, 0, stream>>>(agg, h2, Wl3, Wr3, b3, deg, outp,
                                                         n_tiles, 0);
}